// listNet_32109175505031
// MI455X (gfx1250) — compile-verified
//
#include <hip/hip_runtime.h>
#include <hip/hip_bf16.h>
#include <stdint.h>

// Batched Plackett-Luce / ListNet loss: B = 2^21 rows, K = 11.
// loss(row) = K*max + sum_t log(suffix_cumsum_t(exp(y-max))) - sum_t y_t
// Memory-bound (~101 MB @ 23.3 TB/s ~= 4.3 us). Async global->LDS staging
// (CDNA5 GLOBAL_LOAD_ASYNC_TO_LDS_B128, ASYNCcnt) gives fully coalesced
// 128B traffic despite the 44-byte row stride.

#define KLEN 11
#define THREADS 256
#define ROWS_PER_BLOCK 1024                       // tile = 1024*11*4 = 45056 B
#define TILE_BYTES (ROWS_PER_BLOCK * KLEN * 4)
#define ROUNDS (TILE_BYTES / (THREADS * 16))      // 45056 / 4096 = 11 exact

__global__ __launch_bounds__(THREADS) void listnet_plackett_luce_kernel(
    const float* __restrict__ y, float* __restrict__ out, int nRows) {
  __shared__ float smem[ROWS_PER_BLOCK * KLEN];   // 45056 bytes

  const unsigned tid = threadIdx.x;
  const char* gbase = (const char*)y + (size_t)blockIdx.x * TILE_BYTES;

  // Low 32 bits of the shared-aperture generic address == wave-relative LDS
  // byte offset (aperture tag lives entirely in addr[63:32] on CDNA5).
  const unsigned lbase = (unsigned)(uintptr_t)(void*)&smem[0];

  // Stage the whole tile: 11 rounds of 256 lanes x 16B async copies.
#pragma unroll
  for (int r = 0; r < ROUNDS; ++r) {
    const unsigned off = (unsigned)(r * THREADS * 16) + tid * 16u;
    const unsigned ldsa = lbase + off;
    const char* g = gbase + off;
    asm volatile("global_load_async_to_lds_b128 %0, %1, off"
                 :: "v"(ldsa), "v"(g)
                 : "memory");
  }
  asm volatile("s_wait_asynccnt 0" ::: "memory");
  __syncthreads();

  const size_t rowBase = (size_t)blockIdx.x * ROWS_PER_BLOCK;

#pragma unroll
  for (int j = 0; j < ROWS_PER_BLOCK / THREADS; ++j) {
    const unsigned row = tid + j * THREADS;   // stride-11 float reads: no bank conflicts
    const float* rp = &smem[row * KLEN];

    float yv[KLEN];
#pragma unroll
    for (int k = 0; k < KLEN; ++k) yv[k] = rp[k];

    float m = yv[0];
#pragma unroll
    for (int k = 1; k < KLEN; ++k) m = fmaxf(m, yv[k]);

    float sumy = 0.0f;
#pragma unroll
    for (int k = 0; k < KLEN; ++k) sumy += yv[k];

    float ex[KLEN];
#pragma unroll
    for (int k = 0; k < KLEN; ++k) ex[k] = __expf(yv[k] - m);

    // suffix cumulative sums csum[t] = sum_{l>=t} ex[l]
    float c[KLEN];
    c[KLEN - 1] = ex[KLEN - 1];
#pragma unroll
    for (int k = KLEN - 2; k >= 0; --k) c[k] = c[k + 1] + ex[k];

    // sum_t log(c[t]) == log(prod c[t]); 3 grouped products keep f32 in range
    // (each c[t] >= ex[t] >= exp(-(max-min)); 4-factor products stay normal).
    const float p0 = ((c[0] * c[1]) * (c[2] * c[3]));
    const float p1 = ((c[4] * c[5]) * (c[6] * c[7]));
    const float p2 = ((c[8] * c[9]) * c[10]);
    const float loss = __logf(p0) + __logf(p1) + __logf(p2)
                     + (float)KLEN * m - sumy;

    const size_t orow = rowBase + row;
    if (orow < (size_t)nRows) out[orow] = loss;
  }
}

extern "C" void kernel_launch(void* const* d_in, const int* in_sizes, int n_in,
                              void* d_out, int out_size, void* d_ws, size_t ws_size,
                              hipStream_t stream) {
  (void)n_in; (void)d_ws; (void)ws_size; (void)out_size;
  const float* y = (const float*)d_in[0];
  float* out = (float*)d_out;
  const int total = in_sizes[0];          // B * K
  const int nRows = total / KLEN;         // 2097152
  const int blocks = (nRows + ROWS_PER_BLOCK - 1) / ROWS_PER_BLOCK;  // 2048
  listnet_plackett_luce_kernel<<<blocks, THREADS, 0, stream>>>(y, out, nRows);
}